// GCNConvModel_71588514889832
// MI455X (gfx1250) — compile-verified
//
#include <hip/hip_runtime.h>
#include <math.h>

#define NN   100000
#define NE   1600000
#define INC  1024
#define HID  128
#define OUTC 3
#define KC   64           // K chunk staged in LDS for GEMM1
#define MT   (NN / 16)    // 6250 M-tiles, NN % 16 == 0

// LDS pair layout for GEMM1 B fragments:
//   pair slot = kq*GS + kh*KH_OFF + n   holds (W[4kq+2kh][n], W[4kq+2kh+1][n])
//   16-pair skew between kh halves -> disjoint LDS bank sets for lanes 0-15 / 16-31
#define GS     (2 * HID + 32)   // 288 pairs per k-group
#define KH_OFF (HID + 16)       // 144

typedef __attribute__((ext_vector_type(2))) float v2f;
typedef __attribute__((ext_vector_type(8))) float v8f;

__device__ __forceinline__ void atomic_add_f32(float* p, float v) {
    (void)__hip_atomic_fetch_add(p, v, __ATOMIC_RELAXED, __HIP_MEMORY_SCOPE_AGENT);
}

// ---------------- degree / normalization ----------------
__global__ void k_deg_init(float* deg) {
    int i = blockIdx.x * blockDim.x + threadIdx.x;
    if (i < NN) deg[i] = 1.0f;                       // self-loop contribution
}

__global__ void k_deg_count(const int* __restrict__ dst, float* deg) {
    int e = blockIdx.x * blockDim.x + threadIdx.x;
    if (e < NE) atomic_add_f32(&deg[dst[e]], 1.0f);
}

__global__ void k_dis(float* degdis) {               // deg -> rsqrt(deg), in place
    int i = blockIdx.x * blockDim.x + threadIdx.x;
    if (i < NN) degdis[i] = rsqrtf(degdis[i]);
}

// ---------------- GEMM1: xw = X[NN,1024] @ W1[1024,128] (fp32 WMMA) ----------------
__global__ __launch_bounds__(256) void k_gemm1(const float* __restrict__ X,
                                               const float* __restrict__ W,
                                               float* __restrict__ Y) {
    __shared__ v2f Wl[(KC / 4) * GS];                // 16 * 288 * 8B = 36 KB
    const int tid  = threadIdx.x;
    const int lane = tid & 31;
    const int wv   = tid >> 5;
    const int mt   = blockIdx.x * 8 + wv;
    const bool act = (mt < MT);
    const int kh   = lane >> 4;                      // K-half select (ISA 16x4 f32 A layout)
    const int nc   = lane & 15;                      // A row / B,C,D column
    const float* xr = X + (size_t)(mt * 16 + nc) * INC;

    v8f acc[8];
#pragma unroll
    for (int t = 0; t < 8; ++t)
        acc[t] = (v8f){0.f, 0.f, 0.f, 0.f, 0.f, 0.f, 0.f, 0.f};

    for (int kc = 0; kc < INC; kc += KC) {
        // stage W1 K-chunk into LDS as swizzled K-pairs: (KC/4)*2*HID = 4096 pairs
#pragma unroll
        for (int it = 0; it < 4096 / 256; ++it) {
            int i   = tid + it * 256;
            int kq  = i >> 8;                        // k-group of 4
            int rem = i & 255;
            int kh2 = rem >> 7;
            int n   = rem & 127;
            int k   = kc + kq * 4 + 2 * kh2;
            v2f p;
            p.x = W[(size_t)k * HID + n];            // coalesced across n
            p.y = W[(size_t)(k + 1) * HID + n];
            Wl[kq * GS + kh2 * KH_OFF + n] = p;      // single ds_store_b64
        }
        __syncthreads();

        if (act) {
            if (kc + KC < INC)
                __builtin_prefetch(xr + kc + KC + 2 * kh, 0, 0);
#pragma unroll 4
            for (int kk = 0; kk < KC; kk += 4) {
                // A frag: lanes 0-15 -> K = kk,kk+1 ; lanes 16-31 -> K = kk+2,kk+3
                const v2f a = *(const v2f*)(xr + kc + kk + 2 * kh);
                const int kq = kk >> 2;
#pragma unroll
                for (int t = 0; t < 8; ++t) {
                    // B frag: one aligned ds_load_b64 straight into a VGPR pair
                    const v2f b = Wl[kq * GS + kh * KH_OFF + t * 16 + nc];
                    acc[t] = __builtin_amdgcn_wmma_f32_16x16x4_f32(
                        false, a, false, b, (short)0, acc[t], false, false);
                }
            }
        }
        __syncthreads();
    }

    if (act) {
        // C/D layout: row = v + 8*kh, col = nc
        float* yb = Y + (size_t)(mt * 16 + 8 * kh) * HID + nc;
#pragma unroll
        for (int t = 0; t < 8; ++t)
#pragma unroll
            for (int v = 0; v < 8; ++v)
                yb[(size_t)v * HID + t * 16] = acc[t][v];
    }
}

// ---------------- h = xw * dis^2 + b1  (self-loop term + bias) ----------------
__global__ void k_init_h(const float* __restrict__ xw, const float* __restrict__ dis,
                         const float* __restrict__ b1, float* __restrict__ h) {
    int i = blockIdx.x * blockDim.x + threadIdx.x;
    if (i < NN * HID) {
        int node = i >> 7, c = i & (HID - 1);
        float d = dis[node];
        h[i] = xw[i] * d * d + b1[c];
    }
}

// ---------------- scatter-add layer 1: one wave per edge, 4 ch per lane ----------------
__global__ __launch_bounds__(256) void k_scatter1(const int* __restrict__ src,
                                                  const int* __restrict__ dst,
                                                  const float* __restrict__ dis,
                                                  const float* __restrict__ xw,
                                                  float* __restrict__ h) {
    int e    = (blockIdx.x * 256 + threadIdx.x) >> 5;
    int lane = threadIdx.x & 31;
    if (e >= NE) return;
    int s = src[e], d = dst[e];
    float nrm = dis[s] * dis[d];
    const float4 v = *(const float4*)(xw + (size_t)s * HID + lane * 4);
    float* hp = h + (size_t)d * HID + lane * 4;
    atomic_add_f32(hp + 0, v.x * nrm);
    atomic_add_f32(hp + 1, v.y * nrm);
    atomic_add_f32(hp + 2, v.z * nrm);
    atomic_add_f32(hp + 3, v.w * nrm);
}

__global__ void k_relu(float* h) {
    int i = blockIdx.x * blockDim.x + threadIdx.x;
    if (i < NN * HID) h[i] = fmaxf(h[i], 0.0f);
}

// ---------------- GEMM2: hw2 = h[NN,128] @ W2[128,3] (N padded to 16, fp32 WMMA) ----------------
__global__ __launch_bounds__(256) void k_gemm2(const float* __restrict__ H,
                                               const float* __restrict__ W2,
                                               float* __restrict__ Y) {
    // W2 zero-padded to [128][16], stored as K-pairs: 32 k-groups * 32 pairs = 8 KB
    __shared__ v2f Wl[(HID / 4) * 32];
    const int tid = threadIdx.x;
    for (int i = tid; i < (HID / 4) * 32; i += 256) {
        int kq  = i >> 5;
        int rem = i & 31;
        int kh2 = rem >> 4;
        int n   = rem & 15;
        int k   = kq * 4 + 2 * kh2;
        v2f p;
        p.x = (n < OUTC) ? W2[k * OUTC + n] : 0.0f;
        p.y = (n < OUTC) ? W2[(k + 1) * OUTC + n] : 0.0f;
        Wl[kq * 32 + kh2 * 16 + n] = p;
    }
    __syncthreads();

    const int lane = tid & 31, wv = tid >> 5;
    const int mt = blockIdx.x * 8 + wv;
    if (mt >= MT) return;
    const int kh = lane >> 4, nc = lane & 15;
    const float* hr = H + (size_t)(mt * 16 + nc) * HID;

    v8f acc = (v8f){0.f, 0.f, 0.f, 0.f, 0.f, 0.f, 0.f, 0.f};
#pragma unroll
    for (int kk = 0; kk < HID; kk += 4) {
        const v2f a = *(const v2f*)(hr + kk + 2 * kh);
        const v2f b = Wl[(kk >> 2) * 32 + kh * 16 + nc];
        acc = __builtin_amdgcn_wmma_f32_16x16x4_f32(
            false, a, false, b, (short)0, acc, false, false);
    }

    if (nc < OUTC) {
        float* yb = Y + (size_t)(mt * 16 + 8 * kh) * OUTC + nc;
#pragma unroll
        for (int v = 0; v < 8; ++v)
            yb[(size_t)v * OUTC] = acc[v];
    }
}

// ---------------- out = hw2 * dis^2 + b2 ----------------
__global__ void k_init_out(const float* __restrict__ hw2, const float* __restrict__ dis,
                           const float* __restrict__ b2, float* __restrict__ out) {
    int i = blockIdx.x * blockDim.x + threadIdx.x;
    if (i < NN * OUTC) {
        int node = i / OUTC, c = i - node * OUTC;
        float d = dis[node];
        out[i] = hw2[i] * d * d + b2[c];
    }
}

// ---------------- scatter-add layer 2: one thread per edge, 3 channels ----------------
__global__ void k_scatter2(const int* __restrict__ src, const int* __restrict__ dst,
                           const float* __restrict__ dis, const float* __restrict__ hw2,
                           float* __restrict__ out) {
    int e = blockIdx.x * blockDim.x + threadIdx.x;
    if (e >= NE) return;
    int s = src[e], d = dst[e];
    float nrm = dis[s] * dis[d];
    const float* hp = hw2 + (size_t)s * OUTC;
    float* op = out + (size_t)d * OUTC;
    atomic_add_f32(op + 0, hp[0] * nrm);
    atomic_add_f32(op + 1, hp[1] * nrm);
    atomic_add_f32(op + 2, hp[2] * nrm);
}

extern "C" void kernel_launch(void* const* d_in, const int* in_sizes, int n_in,
                              void* d_out, int out_size, void* d_ws, size_t ws_size,
                              hipStream_t stream) {
    const float* X   = (const float*)d_in[0];
    const int* edges = (const int*)d_in[1];
    const int* src   = edges;                        // edges[0, :]
    const int* dst   = edges + NE;                   // edges[1, :]
    const float* W1  = (const float*)d_in[4];
    const float* b1  = (const float*)d_in[5];
    const float* W2  = (const float*)d_in[6];
    const float* b2  = (const float*)d_in[7];
    float* out = (float*)d_out;

    float* xw  = (float*)d_ws;                       // [NN,128]  51.2 MB
    float* h   = xw  + (size_t)NN * HID;             // [NN,128]  51.2 MB
    float* hw2 = h   + (size_t)NN * HID;             // [NN,3]     1.2 MB
    float* deg = hw2 + (size_t)NN * OUTC;            // [NN] deg -> dis in place

    const int B = 256;
    k_deg_init <<<(NN + B - 1) / B, B, 0, stream>>>(deg);
    k_deg_count<<<(NE + B - 1) / B, B, 0, stream>>>(dst, deg);
    k_dis      <<<(NN + B - 1) / B, B, 0, stream>>>(deg);

    k_gemm1    <<<(MT + 7) / 8, B, 0, stream>>>(X, W1, xw);
    k_init_h   <<<(NN * HID + B - 1) / B, B, 0, stream>>>(xw, deg, b1, h);
    k_scatter1 <<<NE / 8, B, 0, stream>>>(src, dst, deg, xw, h);   // NE waves
    k_relu     <<<(NN * HID + B - 1) / B, B, 0, stream>>>(h);

    k_gemm2    <<<(MT + 7) / 8, B, 0, stream>>>(h, W2, hw2);
    k_init_out <<<(NN * OUTC + B - 1) / B, B, 0, stream>>>(hw2, deg, b2, out);
    k_scatter2 <<<(NE + B - 1) / B, B, 0, stream>>>(src, dst, deg, hw2, out);
}